// SpatialOmicsModel_8289286881686
// MI455X (gfx1250) — compile-verified
//
#include <hip/hip_runtime.h>
#include <math.h>

// ---------------------------------------------------------------------------
// SpatialOmics GAT model for MI455X (gfx1250, wave32).
// Dense projections use V_WMMA_F32_16X16X4_F32 (fp32 matrix pipe):
//   - A tile staged through LDS (ds_load_2addr_b64 in the K-loop)
//   - B pre-packed into K-pair-interleaved layout (one global_load_b64/WMMA)
// Edge softmax / scatter phases are memory/atomic bound: coalesced gathers,
// per-(edge,head) alpha computed once and broadcast via LDS.
// ---------------------------------------------------------------------------

typedef __attribute__((ext_vector_type(2))) float v2f;
typedef __attribute__((ext_vector_type(8))) float v8f;

#define NEG_SLOPE 0.2f
#define HID 64
#define NUM_GRAPHS 64
#define OUT_DIM 32

// ---- ordered-uint encoding so we can do float atomic max via atomicMax(u32)
__device__ __forceinline__ unsigned f2ord(float f) {
  unsigned u = __float_as_uint(f);
  return (u & 0x80000000u) ? ~u : (u | 0x80000000u);
}
__device__ __forceinline__ float ord2f(unsigned u) {
  unsigned v = (u & 0x80000000u) ? (u & 0x7FFFFFFFu) : ~u;
  return __uint_as_float(v);
}
#define ORD_NEG_INF 0x007FFFFFu  // f2ord(-inf)

__device__ __forceinline__ float lrelu(float x) { return x < 0.f ? NEG_SLOPE * x : x; }

// ---------------------------------------------------------------------------
// init helpers
// ---------------------------------------------------------------------------
__global__ void fill_f(float* p, float v, int n) {
  int i = blockIdx.x * blockDim.x + threadIdx.x;
  if (i < n) p[i] = v;
}
__global__ void fill_u(unsigned* p, unsigned v, int n) {
  int i = blockIdx.x * blockDim.x + threadIdx.x;
  if (i < n) p[i] = v;
}

// ---------------------------------------------------------------------------
// pack W [K,M] -> Wt [P][M] of float2 pairs {W[2p][c], W[2p+1][c]}, zero-pad
// rows >= K.  One global_load_b64 then feeds the WMMA B operand directly.
// ---------------------------------------------------------------------------
__global__ void pack_w(const float* __restrict__ W, float* __restrict__ Wt,
                       int K, int M, int P) {
  int i = blockIdx.x * blockDim.x + threadIdx.x;
  if (i >= P * M) return;
  int p = i / M, c = i - p * M;
  int k0 = 2 * p, k1 = 2 * p + 1;
  Wt[2 * i]     = (k0 < K) ? W[k0 * M + c] : 0.f;
  Wt[2 * i + 1] = (k1 < K) ? W[k1 * M + c] : 0.f;
}

// ---------------------------------------------------------------------------
// GEMM 1: xp = concat(protein, spatial, lri) @ W1   [N,132pad] x [132pad,128]
// Block = 256 threads (8 waves); each wave owns one 16x16 column tile.
// A tile (16 x 132, zero-padded) staged in LDS; B from packed Wt1 (66 pairs).
// N is a multiple of 16 (50000 = 3125*16).
// ---------------------------------------------------------------------------
#define KPAD 132

__device__ __forceinline__ float loadX(const float* pe, const float* sp,
                                       const float* lri, int row, int k) {
  if (k < 64)  return pe[row * 64 + k];
  if (k < 66)  return sp[row * 2 + (k - 64)];
  if (k < 130) return lri[row * 64 + (k - 66)];
  return 0.f;
}

__global__ void wmma_gemm_concat(const float* __restrict__ pe,
                                 const float* __restrict__ sp,
                                 const float* __restrict__ lri,
                                 const float* __restrict__ Wt1,  // packed [66][128] x2
                                 float* __restrict__ D,          // [N,128]
                                 int N) {
  __shared__ float As[16 * KPAD];

  const int tid  = threadIdx.x;
  const int row0 = blockIdx.x * 16;

  // ---- stage A tile (zero-padded to K=132) into LDS ----
  for (int idx = tid; idx < 16 * KPAD; idx += 256) {
    int r = idx / KPAD;
    int k = idx - r * KPAD;
    As[idx] = loadX(pe, sp, lri, row0 + r, k);
  }
  __syncthreads();

  const int lane = tid & 31;
  const int wv   = tid >> 5;         // 0..7 -> column tile
  const int col0 = wv * 16;
  const int m    = lane & 15;
  const int kb   = (lane >> 4) << 1; // 0 or 2
  const int col  = col0 + m;
  const float* Arow = As + m * KPAD + kb;
  const v2f*   Bv   = (const v2f*)Wt1 + (size_t)(kb >> 1) * 128 + col;

  v8f acc = {};
#pragma unroll
  for (int k = 0; k < KPAD; k += 4) {
    v2f a, b;
    a.x = Arow[k];
    a.y = Arow[k + 1];
    b   = Bv[(k >> 1) * 128];
    acc = __builtin_amdgcn_wmma_f32_16x16x4_f32(false, a, false, b,
                                                (short)0, acc, false, false);
  }

  const int half = lane >> 4;
  const int n    = lane & 15;
#pragma unroll
  for (int v = 0; v < 8; ++v) {
    int r = row0 + v + 8 * half;
    D[(size_t)r * 128 + col0 + n] = acc[v];
  }
}

// ---------------------------------------------------------------------------
// GEMM 2 (fixed shape): D[N,64] = A[N,128] @ B[128,64]
// Block = 128 threads (4 waves); A tile staged in LDS; packed B (64 pairs).
// ---------------------------------------------------------------------------
__global__ void wmma_gemm_128x64(const float* __restrict__ A,
                                 const float* __restrict__ Wt2,  // packed [64][64] x2
                                 float* __restrict__ D, int N) {
  __shared__ float As[16 * 128];

  const int tid  = threadIdx.x;
  const int row0 = blockIdx.x * 16;

  for (int idx = tid; idx < 16 * 128; idx += 128) {
    int r = idx >> 7;
    int k = idx & 127;
    As[idx] = A[(size_t)(row0 + r) * 128 + k];
  }
  __syncthreads();

  const int lane = tid & 31;
  const int wv   = tid >> 5;         // 0..3 -> column tile
  const int col0 = wv * 16;
  const int m    = lane & 15;
  const int kb   = (lane >> 4) << 1;
  const int col  = col0 + m;
  const float* Arow = As + m * 128 + kb;
  const v2f*   Bv   = (const v2f*)Wt2 + (size_t)(kb >> 1) * 64 + col;

  v8f acc = {};
#pragma unroll
  for (int k = 0; k < 128; k += 4) {
    v2f a, b;
    a.x = Arow[k];
    a.y = Arow[k + 1];
    b   = Bv[(k >> 1) * 64];
    acc = __builtin_amdgcn_wmma_f32_16x16x4_f32(false, a, false, b,
                                                (short)0, acc, false, false);
  }

  const int half = lane >> 4;
  const int n    = lane & 15;
#pragma unroll
  for (int v = 0; v < 8; ++v) {
    int r = row0 + v + 8 * half;
    D[(size_t)r * 64 + col0 + n] = acc[v];
  }
}

// ---------------------------------------------------------------------------
// attention logits: al_src[n,h] = sum_c xp[n,h,c]*a_src[h,c]; same for dst
// ---------------------------------------------------------------------------
template <int H, int C>
__global__ void att_logits(const float* __restrict__ xp,
                           const float* __restrict__ a_src,
                           const float* __restrict__ a_dst,
                           float* __restrict__ al_s, float* __restrict__ al_d,
                           int N) {
  int idx = blockIdx.x * blockDim.x + threadIdx.x;
  if (idx >= N * H) return;
  int n = idx / H, h = idx - (idx / H) * H;
  const float* xr = xp + (size_t)n * H * C + h * C;
  const float* as = a_src + h * C;
  const float* ad = a_dst + h * C;
  float ss = 0.f, sd = 0.f;
#pragma unroll 8
  for (int c = 0; c < C; ++c) {
    float v = xr[c];
    ss += v * as[c];
    sd += v * ad[c];
  }
  al_s[idx] = ss;
  al_d[idx] = sd;
}

// ---------------------------------------------------------------------------
// edge pass 1: segment max of leaky_relu(al_src[s]+al_dst[d]) over dst
// ---------------------------------------------------------------------------
template <int H>
__global__ void edge_max(const int* __restrict__ ei, int E, int N,
                         const float* __restrict__ al_s,
                         const float* __restrict__ al_d,
                         unsigned* __restrict__ m_enc) {
  int i = blockIdx.x * blockDim.x + threadIdx.x;
  if (i >= E + N) return;
  int s = (i < E) ? ei[i] : (i - E);
  int d = (i < E) ? ei[E + i] : (i - E);
#pragma unroll
  for (int h = 0; h < H; ++h) {
    float e = lrelu(al_s[s * H + h] + al_d[d * H + h]);
    atomicMax(&m_enc[d * H + h], f2ord(e));
  }
}

// ---------------------------------------------------------------------------
// edge pass 2: denom[d,h] += exp(e - m[d,h])
// ---------------------------------------------------------------------------
template <int H>
__global__ void edge_denom(const int* __restrict__ ei, int E, int N,
                           const float* __restrict__ al_s,
                           const float* __restrict__ al_d,
                           const unsigned* __restrict__ m_enc,
                           float* __restrict__ denom) {
  int i = blockIdx.x * blockDim.x + threadIdx.x;
  if (i >= E + N) return;
  int s = (i < E) ? ei[i] : (i - E);
  int d = (i < E) ? ei[E + i] : (i - E);
#pragma unroll
  for (int h = 0; h < H; ++h) {
    float e = lrelu(al_s[s * H + h] + al_d[d * H + h]);
    float ex = expf(e - ord2f(m_enc[d * H + h]));
    atomicAdd(&denom[d * H + h], ex);
  }
}

// ---------------------------------------------------------------------------
// edge pass 3: out[d, h*C+c] += alpha * xp[s, h*C+c]
// blockDim = 256 -> EPB = 256/(H*C) edges per block.  alpha is computed once
// per (edge, head) by the first EPB*H threads and broadcast through LDS
// (removes ~H*C redundant expf per edge).
// ---------------------------------------------------------------------------
template <int H, int C>
__global__ void edge_scatter(const int* __restrict__ ei, int E, int N,
                             const float* __restrict__ xp,
                             const float* __restrict__ al_s,
                             const float* __restrict__ al_d,
                             const unsigned* __restrict__ m_enc,
                             const float* __restrict__ denom,
                             float* __restrict__ out) {
  constexpr int HC  = H * C;
  constexpr int EPB = 256 / HC;
  __shared__ float s_alpha[EPB * H];

  const int tid  = threadIdx.x;
  const int le   = tid / HC;          // local edge 0..EPB-1 (shift: HC pow2)
  const int c    = tid - le * HC;
  const int Etot = E + N;

  if (tid < EPB * H) {
    int lee = tid / H, hh = tid - (tid / H) * H;
    int ii  = blockIdx.x * EPB + lee;
    float a = 0.f;
    if (ii < Etot) {
      int ss = (ii < E) ? ei[ii] : (ii - E);
      int dd = (ii < E) ? ei[E + ii] : (ii - E);
      float e  = lrelu(al_s[ss * H + hh] + al_d[dd * H + hh]);
      float ex = expf(e - ord2f(m_enc[dd * H + hh]));
      a = ex / denom[dd * H + hh];
    }
    s_alpha[tid] = a;
  }
  __syncthreads();

  const int i = blockIdx.x * EPB + le;
  if (i >= Etot) return;
  int s = (i < E) ? ei[i] : (i - E);
  int d = (i < E) ? ei[E + i] : (i - E);
  const int h = c / C;
  float alpha = s_alpha[le * H + h];
  atomicAdd(&out[(size_t)d * HC + c], alpha * xp[(size_t)s * HC + c]);
}

// ---------------------------------------------------------------------------
// bias add in-place: x[n, j] += b[j]
// ---------------------------------------------------------------------------
__global__ void bias_add(float* __restrict__ x, const float* __restrict__ b,
                         int N, int Dd) {
  int i = blockIdx.x * blockDim.x + threadIdx.x;
  if (i < N * Dd) x[i] += b[i % Dd];
}

// ---------------------------------------------------------------------------
// global mean pool (accumulate phase)
// ---------------------------------------------------------------------------
__global__ void pool_acc(const float* __restrict__ h2,
                         const int* __restrict__ batch,
                         float* __restrict__ pooled, float* __restrict__ counts,
                         int N) {
  int i = blockIdx.x * blockDim.x + threadIdx.x;
  if (i >= N * HID) return;
  int n = i >> 6, c = i & 63;
  int g = batch[n];
  atomicAdd(&pooled[g * HID + c], h2[i]);
  if (c == 0) atomicAdd(&counts[g], 1.0f);
}

__global__ void pool_div(float* __restrict__ pooled,
                         const float* __restrict__ counts) {
  int i = blockIdx.x * blockDim.x + threadIdx.x;
  if (i >= NUM_GRAPHS * HID) return;
  int g = i >> 6;
  pooled[i] /= fmaxf(counts[g], 1.0f);
}

// ---------------------------------------------------------------------------
// decoder MLP (tiny; scalar)
// ---------------------------------------------------------------------------
__global__ void dec1(const float* __restrict__ pooled,
                     const float* __restrict__ Wd1, const float* __restrict__ bd1,
                     float* __restrict__ d1) {
  int i = blockIdx.x * blockDim.x + threadIdx.x;
  if (i >= NUM_GRAPHS * HID) return;
  int g = i >> 6, j = i & 63;
  float s = bd1[j];
  for (int k = 0; k < HID; ++k) s += pooled[g * HID + k] * Wd1[k * HID + j];
  d1[i] = s > 0.f ? s : 0.f;
}

__global__ void dec2(const float* __restrict__ d1,
                     const float* __restrict__ Wd2, const float* __restrict__ bd2,
                     float* __restrict__ outp) {
  int i = blockIdx.x * blockDim.x + threadIdx.x;
  if (i >= NUM_GRAPHS * OUT_DIM) return;
  int g = i >> 5, o = i & 31;
  float s = bd2[o];
  for (int k = 0; k < HID; ++k) s += d1[g * HID + k] * Wd2[k * OUT_DIM + o];
  outp[i] = s;
}

// ---------------------------------------------------------------------------
// host launcher
// ---------------------------------------------------------------------------
extern "C" void kernel_launch(void* const* d_in, const int* in_sizes, int n_in,
                              void* d_out, int out_size, void* d_ws, size_t ws_size,
                              hipStream_t stream) {
  const float* pe     = (const float*)d_in[0];
  const float* sp     = (const float*)d_in[1];
  const float* lri    = (const float*)d_in[2];
  const int*   ei     = (const int*)d_in[3];
  const int*   batch  = (const int*)d_in[4];
  const float* W1     = (const float*)d_in[5];
  const float* a_src1 = (const float*)d_in[6];
  const float* a_dst1 = (const float*)d_in[7];
  const float* b1     = (const float*)d_in[8];
  const float* W2     = (const float*)d_in[9];
  const float* a_src2 = (const float*)d_in[10];
  const float* a_dst2 = (const float*)d_in[11];
  const float* b2     = (const float*)d_in[12];
  const float* Wd1    = (const float*)d_in[13];
  const float* bd1    = (const float*)d_in[14];
  const float* Wd2    = (const float*)d_in[15];
  const float* bd2    = (const float*)d_in[16];
  float* outp = (float*)d_out;

  const int N = in_sizes[0] / 64;   // 50000 (multiple of 16)
  const int E = in_sizes[3] / 2;    // 1600000
  const int Etot = E + N;           // with self-loops

  // ---- workspace layout (floats) ----
  float* ws = (float*)d_ws;
  size_t o = 0;
  float*    xp1   = ws + o; o += (size_t)N * 128;
  float*    h1    = ws + o; o += (size_t)N * 128;
  float*    xp2   = ws + o; o += (size_t)N * 64;
  float*    h2    = ws + o; o += (size_t)N * 64;
  float*    al_s1 = ws + o; o += (size_t)N * 2;
  float*    al_d1 = ws + o; o += (size_t)N * 2;
  unsigned* m1    = (unsigned*)(ws + o); o += (size_t)N * 2;
  float*    den1  = ws + o; o += (size_t)N * 2;
  float*    al_s2 = ws + o; o += (size_t)N;
  float*    al_d2 = ws + o; o += (size_t)N;
  unsigned* m2    = (unsigned*)(ws + o); o += (size_t)N;
  float*    den2  = ws + o; o += (size_t)N;
  float*    pooled = ws + o; o += NUM_GRAPHS * HID;
  float*    counts = ws + o; o += NUM_GRAPHS;
  float*    d1buf  = ws + o; o += NUM_GRAPHS * HID;
  float*    Wt1    = ws + o; o += (size_t)(KPAD / 2) * 128 * 2;  // 66 pairs x 128
  float*    Wt2    = ws + o; o += (size_t)64 * 64 * 2;           // 64 pairs x 64
  (void)ws_size; (void)n_in; (void)out_size;

  const int TB = 256;
  auto blocks = [](long long n, int tb) { return (unsigned)((n + tb - 1) / tb); };

  // ---- init scratch (scatter targets & softmax state) + weight packing ----
  fill_f<<<blocks((long long)N * 128, TB), TB, 0, stream>>>(h1, 0.f, N * 128);
  fill_f<<<blocks((long long)N * 64,  TB), TB, 0, stream>>>(h2, 0.f, N * 64);
  fill_u<<<blocks((long long)N * 2,   TB), TB, 0, stream>>>(m1, ORD_NEG_INF, N * 2);
  fill_f<<<blocks((long long)N * 2,   TB), TB, 0, stream>>>(den1, 0.f, N * 2);
  fill_u<<<blocks(N, TB), TB, 0, stream>>>(m2, ORD_NEG_INF, N);
  fill_f<<<blocks(N, TB), TB, 0, stream>>>(den2, 0.f, N);
  fill_f<<<blocks(NUM_GRAPHS * HID, TB), TB, 0, stream>>>(pooled, 0.f, NUM_GRAPHS * HID);
  fill_f<<<blocks(NUM_GRAPHS, TB), TB, 0, stream>>>(counts, 0.f, NUM_GRAPHS);
  pack_w<<<blocks((KPAD / 2) * 128, TB), TB, 0, stream>>>(W1, Wt1, 130, 128, KPAD / 2);
  pack_w<<<blocks(64 * 64, TB), TB, 0, stream>>>(W2, Wt2, 128, 64, 64);

  // ================= layer 1: GATConv(130 -> 64, heads=2) =================
  wmma_gemm_concat<<<N / 16, 8 * 32, 0, stream>>>(pe, sp, lri, Wt1, xp1, N);
  att_logits<2, HID><<<blocks((long long)N * 2, TB), TB, 0, stream>>>(
      xp1, a_src1, a_dst1, al_s1, al_d1, N);
  edge_max<2><<<blocks(Etot, TB), TB, 0, stream>>>(ei, E, N, al_s1, al_d1, m1);
  edge_denom<2><<<blocks(Etot, TB), TB, 0, stream>>>(ei, E, N, al_s1, al_d1, m1, den1);
  edge_scatter<2, HID><<<blocks(Etot, 2), 256, 0, stream>>>(
      ei, E, N, xp1, al_s1, al_d1, m1, den1, h1);
  bias_add<<<blocks((long long)N * 128, TB), TB, 0, stream>>>(h1, b1, N, 128);

  // ================= layer 2: GATConv(128 -> 64, heads=1) =================
  wmma_gemm_128x64<<<N / 16, 4 * 32, 0, stream>>>(h1, Wt2, xp2, N);
  att_logits<1, HID><<<blocks(N, TB), TB, 0, stream>>>(xp2, a_src2, a_dst2,
                                                       al_s2, al_d2, N);
  edge_max<1><<<blocks(Etot, TB), TB, 0, stream>>>(ei, E, N, al_s2, al_d2, m2);
  edge_denom<1><<<blocks(Etot, TB), TB, 0, stream>>>(ei, E, N, al_s2, al_d2, m2, den2);
  edge_scatter<1, HID><<<blocks(Etot, 4), 256, 0, stream>>>(
      ei, E, N, xp2, al_s2, al_d2, m2, den2, h2);
  bias_add<<<blocks((long long)N * 64, TB), TB, 0, stream>>>(h2, b2, N, 64);

  // ================= pool + decoder =================
  pool_acc<<<blocks((long long)N * HID, TB), TB, 0, stream>>>(h2, batch, pooled,
                                                              counts, N);
  pool_div<<<blocks(NUM_GRAPHS * HID, TB), TB, 0, stream>>>(pooled, counts);
  dec1<<<blocks(NUM_GRAPHS * HID, TB), TB, 0, stream>>>(pooled, Wd1, bd1, d1buf);
  dec2<<<blocks(NUM_GRAPHS * OUT_DIM, TB), TB, 0, stream>>>(d1buf, Wd2, bd2, outp);
}